// PointerNetwork_89206470737972
// MI455X (gfx1250) — compile-verified
//
#include <hip/hip_runtime.h>
#include <hip/hip_bf16.h>
#include <math.h>

// ---------------------------------------------------------------------------
// Pointer network on gfx1250 (wave32, WMMA).
//   B=256, S=128, H=512, F_IN=1.
// Since F_IN==1, embed+input-proj collapses to a rank-1 update:
//   gates = h @ Whh^T + x[b]*u[n] + w[n],   u = Wih@We[:,0], w = Wih@be + b
// Recurrent GEMMs use V_WMMA_F32_16X16X4_F32 (fp32 A/B, fp32 accum) with a
// 32(M)x64(N) strip per wave and a double-buffered K pipeline so fragment
// loads overlap matrix ops instead of hitting s_wait_loadcnt 0 each chunk.
// ---------------------------------------------------------------------------

typedef float v2f __attribute__((ext_vector_type(2)));
typedef float v8f __attribute__((ext_vector_type(8)));

#define PN_B  256
#define PN_S  128
#define PN_H  512
#define PN_FH 2048   // 4*H

// Fragment loads for one K-chunk (4 wide): 2 A-subtiles, 4 B-subtiles.
__device__ __forceinline__ void load_frags(
    const float* __restrict__ arow0, const float* __restrict__ arow1,
    const float* __restrict__ wrow, int ldw, int k,
    v2f a[2], v2f b[4])
{
  a[0] = *(const v2f*)(arow0 + k);
  a[1] = *(const v2f*)(arow1 + k);
#pragma unroll
  for (int j = 0; j < 4; ++j)
    b[j] = *(const v2f*)(wrow + (size_t)(j << 4) * ldw + k);
}

// 8 WMMAs: 2 M-subtiles x 4 N-subtiles, one K-chunk.
__device__ __forceinline__ void wmma8(const v2f a[2], const v2f b[4], v8f acc[2][4]) {
#pragma unroll
  for (int mi = 0; mi < 2; ++mi)
#pragma unroll
    for (int j = 0; j < 4; ++j)
      acc[mi][j] = __builtin_amdgcn_wmma_f32_16x16x4_f32(
          /*neg_a=*/false, a[mi], /*neg_b=*/false, b[j],
          /*c_mod=*/(short)0, acc[mi][j], /*reuse_a=*/false, /*reuse_b=*/false);
}

// ---------------------------------------------------------------------------
// GEMM core: one wave computes a 32(M) x 64(N) strip of  out = A @ W^T.
// A is (M x K) row-major (lda), W is (N x K) row-major (ldw) => B[k][n]=W[n][k].
// WMMA f32 16x16x4 fragment layout (ISA 7.12.2):
//   A 16x4 : lane l & l+16 hold row M=l; lane<16 K={k0,k0+1}, lane>=16 K={k0+2,k0+3}
//   B 4x16 : symmetric (lane -> column N=lane&15, K pair selected by lane>>4)
//   C/D    : element (m,n): n = lane&15, m = vgpr + 8*(lane>>4)
// K pipeline: chunks k and k+4 live in ping-pong registers; loads for chunk
// k+8 issue before the WMMAs consuming chunk k+4. Last two chunks peeled so
// no load reads past A/W.
// ---------------------------------------------------------------------------
__device__ __forceinline__ void wmma_strip_32x64(
    const float* __restrict__ A, int lda,
    const float* __restrict__ W, int ldw,
    int K, int m0, int n0, v8f acc[2][4])
{
  const int lane = threadIdx.x & 31;
  const int half = lane >> 4;
  const int l    = lane & 15;
  const float* arow0 = A + (size_t)(m0 + l) * lda + 2 * half;
  const float* arow1 = arow0 + (size_t)16 * lda;
  const float* wrow  = W + (size_t)(n0 + l) * ldw + 2 * half;

  v2f a0[2], b0[4], a1[2], b1[4];
  load_frags(arow0, arow1, wrow, ldw, 0, a0, b0);
  load_frags(arow0, arow1, wrow, ldw, 4, a1, b1);

  int k0 = 0;
  for (; k0 + 8 < K; k0 += 8) {
    wmma8(a0, b0, acc);
    load_frags(arow0, arow1, wrow, ldw, k0 + 8, a0, b0);
    wmma8(a1, b1, acc);
    load_frags(arow0, arow1, wrow, ldw, k0 + 12, a1, b1);
  }
  wmma8(a0, b0, acc);   // chunk K-8
  wmma8(a1, b1, acc);   // chunk K-4
}

// gates = A @ W^T + x[m]*u[n] + w[n]      (A:(M,K), W:(N,K), out:(M,N))
__global__ void __launch_bounds__(256) wmma_gates_kernel(
    const float* __restrict__ A, const float* __restrict__ W,
    const float* __restrict__ xv, int xstride,
    const float* __restrict__ uvec, const float* __restrict__ wvec,
    float* __restrict__ out, int N, int K, int lda, int ldw)
{
  const int wid    = blockIdx.x * (blockDim.x >> 5) + (threadIdx.x >> 5);
  const int tilesN = N >> 6;
  const int m0 = (wid / tilesN) << 5;
  const int n0 = (wid % tilesN) << 6;
  const int lane = threadIdx.x & 31;
  const int half = lane >> 4;
  const int l    = lane & 15;

  v8f acc[2][4] = {};
  wmma_strip_32x64(A, lda, W, ldw, K, m0, n0, acc);

  float xr[2][8];
#pragma unroll
  for (int mi = 0; mi < 2; ++mi)
#pragma unroll
    for (int r = 0; r < 8; ++r)
      xr[mi][r] = xv[(size_t)(m0 + (mi << 4) + r + (half << 3)) * xstride];

#pragma unroll
  for (int j = 0; j < 4; ++j) {
    const int n = n0 + (j << 4) + l;
    const float uv = uvec[n];
    const float wv = wvec[n];
#pragma unroll
    for (int mi = 0; mi < 2; ++mi)
#pragma unroll
      for (int r = 0; r < 8; ++r) {
        const int m = m0 + (mi << 4) + r + (half << 3);
        out[(size_t)m * N + n] = acc[mi][j][r] + xr[mi][r] * uv + wv;
      }
  }
}

// out = A @ W^T + bias[n]   (out row stride ldo; used for w1enc[:,t,:] and q)
__global__ void __launch_bounds__(256) wmma_bias_kernel(
    const float* __restrict__ A, const float* __restrict__ W,
    const float* __restrict__ bias,
    float* __restrict__ out, long ldo, int N, int K, int lda, int ldw)
{
  const int wid    = blockIdx.x * (blockDim.x >> 5) + (threadIdx.x >> 5);
  const int tilesN = N >> 6;
  const int m0 = (wid / tilesN) << 5;
  const int n0 = (wid % tilesN) << 6;
  const int lane = threadIdx.x & 31;
  const int half = lane >> 4;
  const int l    = lane & 15;

  v8f acc[2][4] = {};
  wmma_strip_32x64(A, lda, W, ldw, K, m0, n0, acc);

#pragma unroll
  for (int j = 0; j < 4; ++j) {
    const int n = n0 + (j << 4) + l;
    const float bv = bias[n];
#pragma unroll
    for (int mi = 0; mi < 2; ++mi)
#pragma unroll
      for (int r = 0; r < 8; ++r) {
        const int m = m0 + (mi << 4) + r + (half << 3);
        out[(size_t)m * ldo + n] = acc[mi][j][r] + bv;
      }
  }
}

// ---------------------------------------------------------------------------
// u[n] = Wih[n,:] . We[:,0] ;  w[n] = Wih[n,:] . be + b[n]      (n < 4H)
// ---------------------------------------------------------------------------
__global__ void precompute_kernel(const float* __restrict__ Wih,
                                  const float* __restrict__ We,
                                  const float* __restrict__ be,
                                  const float* __restrict__ b,
                                  float* __restrict__ u, float* __restrict__ w)
{
  const int n = blockIdx.x * blockDim.x + threadIdx.x;
  if (n >= PN_FH) return;
  const float* row = Wih + (size_t)n * PN_H;
  float su = 0.f, sw = 0.f;
  for (int k = 0; k < PN_H; ++k) {
    const float wv = row[k];
    su = fmaf(wv, We[k], su);
    sw = fmaf(wv, be[k], sw);
  }
  u[n] = su;
  w[n] = sw + b[n];
}

__global__ void zero_kernel(float* __restrict__ p, int n) {
  const int i = blockIdx.x * blockDim.x + threadIdx.x;
  if (i < n) p[i] = 0.f;
}

// ---------------------------------------------------------------------------
// LSTM pointwise: c = sig(f)*c + sig(i)*tanh(g);  h = sig(o)*tanh(c)
// ---------------------------------------------------------------------------
__global__ void __launch_bounds__(256) lstm_pointwise_kernel(
    const float* __restrict__ gates, float* __restrict__ h, float* __restrict__ c)
{
  const int idx = blockIdx.x * blockDim.x + threadIdx.x;   // b*H + n
  const int b = idx >> 9;            // /H
  const int n = idx & (PN_H - 1);
  const float* g = gates + (size_t)b * PN_FH;
  const float gi = g[n];
  const float gf = g[PN_H + n];
  const float gg = g[2 * PN_H + n];
  const float go = g[3 * PN_H + n];
  const float si = 1.f / (1.f + __expf(-gi));
  const float sf = 1.f / (1.f + __expf(-gf));
  const float so = 1.f / (1.f + __expf(-go));
  const float cn = sf * c[idx] + si * tanhf(gg);
  c[idx] = cn;
  h[idx] = so * tanhf(cn);
}

// ---------------------------------------------------------------------------
// Attention energies: us[b,s] = v . tanh(w1enc[b,s,:] + q[b,:]) + bv
// One wave per (b,s); lane-strided dot + xor-shuffle reduction.
// ---------------------------------------------------------------------------
__global__ void __launch_bounds__(256) attn_score_kernel(
    const float* __restrict__ w1enc, const float* __restrict__ q,
    const float* __restrict__ v, const float* __restrict__ bv,
    float* __restrict__ us)
{
  const int wid  = blockIdx.x * (blockDim.x >> 5) + (threadIdx.x >> 5);
  const int lane = threadIdx.x & 31;
  const int b = wid >> 7;            // /S
  const float* wp = w1enc + (size_t)wid * PN_H;   // (b*S+s)*H
  const float* qp = q + (size_t)b * PN_H;
  float sum = 0.f;
  for (int n = lane; n < PN_H; n += 32)
    sum = fmaf(v[n], tanhf(wp[n] + qp[n]), sum);
#pragma unroll
  for (int off = 16; off > 0; off >>= 1)
    sum += __shfl_xor(sum, off, 32);
  if (lane == 0) us[wid] = sum + bv[0];
}

// ---------------------------------------------------------------------------
// Per-batch log_softmax over S, argmax (first max, matching jnp.argmax),
// write scores row out[b, t, :], pointer out2[b, t], and next decoder input.
// One block of S=128 threads per batch element.
// ---------------------------------------------------------------------------
__global__ void __launch_bounds__(128) softmax_argmax_kernel(
    const float* __restrict__ us, const float* __restrict__ input_seq,
    float* __restrict__ out, float* __restrict__ x_next, int t)
{
  __shared__ float sv[PN_S];
  __shared__ int   si[PN_S];
  __shared__ float se[PN_S];
  const int b = blockIdx.x;
  const int s = threadIdx.x;
  const float val = us[b * PN_S + s];
  sv[s] = val; si[s] = s;
  __syncthreads();
  for (int st = PN_S >> 1; st > 0; st >>= 1) {
    if (s < st) {
      const float o = sv[s + st]; const int oi = si[s + st];
      if (o > sv[s] || (o == sv[s] && oi < si[s])) { sv[s] = o; si[s] = oi; }
    }
    __syncthreads();
  }
  const float mx = sv[0];
  const int  idx = si[0];
  se[s] = __expf(val - mx);
  __syncthreads();
  for (int st = PN_S >> 1; st > 0; st >>= 1) {
    if (s < st) se[s] += se[s + st];
    __syncthreads();
  }
  const float lse = mx + __logf(se[0]);
  // pointers_scores[b, t, s]
  out[((size_t)b * PN_S + t) * PN_S + s] = val - lse;
  if (s == 0) {
    // pointers[b, t] (stored as float after the score block)
    out[(size_t)PN_B * PN_S * PN_S + (size_t)b * PN_S + t] = (float)idx;
    x_next[b] = input_seq[(size_t)b * PN_S + idx];   // input_seq[b, idx, 0]
  }
}

// ---------------------------------------------------------------------------
extern "C" void kernel_launch(void* const* d_in, const int* in_sizes, int n_in,
                              void* d_out, int out_size, void* d_ws, size_t ws_size,
                              hipStream_t stream) {
  (void)in_sizes; (void)n_in; (void)out_size; (void)ws_size;
  const float* input_seq = (const float*)d_in[0];
  const float* We_e  = (const float*)d_in[1];
  const float* be_e  = (const float*)d_in[2];
  const float* Wih_e = (const float*)d_in[3];
  const float* Whh_e = (const float*)d_in[4];
  const float* b_e   = (const float*)d_in[5];
  const float* We_d  = (const float*)d_in[6];
  const float* be_d  = (const float*)d_in[7];
  const float* Wih_d = (const float*)d_in[8];
  const float* Whh_d = (const float*)d_in[9];
  const float* b_d   = (const float*)d_in[10];
  const float* W1    = (const float*)d_in[11];
  const float* b1    = (const float*)d_in[12];
  const float* W2    = (const float*)d_in[13];
  const float* b2    = (const float*)d_in[14];
  const float* v     = (const float*)d_in[15];
  const float* bv    = (const float*)d_in[16];
  float* out = (float*)d_out;

  // workspace carve-up (floats): ~71 MB total
  float* ws    = (float*)d_ws;
  float* w1enc = ws;                                   // B*S*H
  float* gates = w1enc + (size_t)PN_B * PN_S * PN_H;   // B*4H
  float* h     = gates + (size_t)PN_B * PN_FH;         // B*H
  float* c     = h + (size_t)PN_B * PN_H;              // B*H
  float* q     = c + (size_t)PN_B * PN_H;              // B*H
  float* us    = q + (size_t)PN_B * PN_H;              // B*S
  float* u_e   = us + (size_t)PN_B * PN_S;             // 4H
  float* w_e   = u_e + PN_FH;                          // 4H
  float* u_d   = w_e + PN_FH;                          // 4H
  float* w_d   = u_d + PN_FH;                          // 4H
  float* xn    = w_d + PN_FH;                          // B

  // rank-1 folded input projections
  precompute_kernel<<<PN_FH / 256, 256, 0, stream>>>(Wih_e, We_e, be_e, b_e, u_e, w_e);
  precompute_kernel<<<PN_FH / 256, 256, 0, stream>>>(Wih_d, We_d, be_d, b_d, u_d, w_d);

  // h = c = 0 (contiguous), x0 = 0
  zero_kernel<<<(2 * PN_B * PN_H) / 256, 256, 0, stream>>>(h, 2 * PN_B * PN_H);
  zero_kernel<<<1, 256, 0, stream>>>(xn, PN_B);

  // grid sizes: 1 wave per 32x64 output strip, 8 waves (256 thr) per block
  const int gates_blocks = (PN_B / 32) * (PN_FH / 64) / 8;   // 32
  const int proj_blocks  = (PN_B / 32) * (PN_H / 64) / 8;    // 8
  const int pw_blocks    = (PN_B * PN_H) / 256;              // 512
  const int attn_blocks  = (PN_B * PN_S) / 8;                // 4096

  // -------- encoder: h@Whh_e^T recurrence + incremental w1enc --------
  for (int t = 0; t < PN_S; ++t) {
    wmma_gates_kernel<<<gates_blocks, 256, 0, stream>>>(
        h, Whh_e, input_seq + t, PN_S, u_e, w_e, gates, PN_FH, PN_H, PN_H, PN_H);
    lstm_pointwise_kernel<<<pw_blocks, 256, 0, stream>>>(gates, h, c);
    // w1enc[:, t, :] = h @ W1^T + b1   (hoisted attention term)
    wmma_bias_kernel<<<proj_blocks, 256, 0, stream>>>(
        h, W1, b1, w1enc + (size_t)t * PN_H, (long)PN_S * PN_H, PN_H, PN_H, PN_H, PN_H);
  }

  // -------- decoder: LSTMCell + additive attention + pointer --------
  for (int t = 0; t < PN_S; ++t) {
    wmma_gates_kernel<<<gates_blocks, 256, 0, stream>>>(
        h, Whh_d, xn, 1, u_d, w_d, gates, PN_FH, PN_H, PN_H, PN_H);
    lstm_pointwise_kernel<<<pw_blocks, 256, 0, stream>>>(gates, h, c);
    wmma_bias_kernel<<<8, 256, 0, stream>>>(
        h, W2, b2, q, (long)PN_H, PN_H, PN_H, PN_H, PN_H);
    attn_score_kernel<<<attn_blocks, 256, 0, stream>>>(w1enc, q, v, bv, us);
    softmax_argmax_kernel<<<PN_B, PN_S, 0, stream>>>(us, input_seq, out, xn, t);
  }
  (void)proj_blocks;
}